// PredictionCapsule_23063974379795
// MI455X (gfx1250) — compile-verified
//
#include <hip/hip_runtime.h>

// Problem constants (from reference setup_inputs)
#define B_TOT     256
#define IN_CAPS   4096
#define IN_DIM    8
#define CAPS      16
#define DIM       16
#define ROUTING_ITERS 3
#define EPS       1e-7f

#define SUB       128           // i sub-chunk staged in LDS
#define XSTRIDE   136           // padded LDS row stride (floats) to kill bank conflicts
#define IBLK      4             // input capsules per barrier phase (u kept in VGPRs)

typedef __attribute__((ext_vector_type(2))) float v2f;
typedef __attribute__((ext_vector_type(8))) float v8f;

// DPP16 rotate-add within each 16-lane row: after 4 steps every lane of the
// row holds the full 16-lane sum. Pure VALU (v_mov_b32_dpp / v_add_f32), no LDS.
#define DPP_ROR_ADD(v, ctrl)                                                   \
  do {                                                                         \
    int _x = __builtin_amdgcn_mov_dpp(__float_as_int(v), (ctrl), 0xF, 0xF, 1); \
    (v) += __int_as_float(_x);                                                 \
  } while (0)

__device__ __forceinline__ float rowsum16(float v) {
  DPP_ROR_ADD(v, 0x128);  // ROW_ROR:8
  DPP_ROR_ADD(v, 0x124);  // ROW_ROR:4
  DPP_ROR_ADD(v, 0x122);  // ROW_ROR:2
  DPP_ROR_ADD(v, 0x121);  // ROW_ROR:1
  return v;
}

__global__ void __launch_bounds__(256) init_pvec(float* __restrict__ pvec) {
  int i = blockIdx.x * blockDim.x + threadIdx.x;
  if (i < B_TOT * CAPS * DIM) pvec[i] = 0.f;
}

// One workgroup = 16 batch rows (M tile) x all 16 capsules (one wave per capsule)
// for one chunk of input capsules. Recomputes u via V_WMMA_F32_16X16X4_F32,
// does logits + softmax + weighted-sum fused, emits partial s per chunk.
__global__ void __launch_bounds__(512) caps_route_accum(
    const float* __restrict__ x, const float* __restrict__ W,
    const float* __restrict__ pvec, float* __restrict__ partial, int NI)
{
  const int ci   = blockIdx.x;            // i-chunk index
  const int bt   = blockIdx.y;            // batch tile (16 rows)
  const int CHUNK = IN_CAPS / NI;
  const int i0   = ci * CHUNK;
  const int tid  = threadIdx.x;
  const int lane = tid & 31;
  const int a    = tid >> 5;              // capsule handled by this wave
  const int half = lane >> 4;
  const int col  = lane & 15;             // = d for B/D layout, = b-row for A layout

  __shared__ __align__(16) float xs[SUB * XSTRIDE];   // x tile [i][b][c] (padded)
  __shared__ float lg[2][IBLK][CAPS][16];             // logits double buffer [t][a][b]

  // Preload routing state Pvec for this wave's (b-rows, a, d=col): constant over i.
  float pv[8];
  #pragma unroll
  for (int j = 0; j < 8; ++j) {
    int bg = bt * 16 + j + 8 * half;
    pv[j] = pvec[(bg * CAPS + a) * DIM + col];
  }

  v8f sacc = {0.f, 0.f, 0.f, 0.f, 0.f, 0.f, 0.f, 0.f};   // s accumulator in D layout
  const float* Wa = W + (size_t)a * IN_CAPS * DIM * IN_DIM;

  int p = 0;
  for (int is = 0; is < CHUNK; is += SUB) {
    __syncthreads();   // protect xs from previous sub-chunk's readers
    // ---- stage x[btile, i0+is .. +SUB, 0..8) into LDS, layout xs[i][b][c] ----
    {
      const int bg = tid >> 5;            // 16 groups of 32 lanes, one batch row each
      const int l  = tid & 31;
      const float* xb = x + (((size_t)(bt * 16 + bg)) * IN_CAPS + (size_t)(i0 + is)) * IN_DIM;
      #pragma unroll
      for (int q = 0; q < 8; ++q) {
        int fo = q * 128 + l * 4;                       // float offset, 16B aligned
        float4 vv = *(const float4*)(xb + fo);
        int il = fo >> 3;                               // i_local
        int c  = fo & 7;                                // 0 or 4
        *(float4*)(xs + il * XSTRIDE + bg * 8 + c) = vv;
      }
    }
    __syncthreads();

    for (int ib = 0; ib < SUB; ib += IBLK) {
      v8f u[IBLK];

      // ---- Phase A: u tiles + logits for IBLK input capsules (u stays in VGPRs)
      #pragma unroll
      for (int t = 0; t < IBLK; ++t) {
        const int ii = ib + t;
        const int ig = i0 + is + ii;

        // A operand (16x4 f32): lanes 0-15 rows b=0..15 K={0,1}; lanes 16-31 K={2,3}
        const float* xrow = xs + ii * XSTRIDE + col * 8;
        v2f A0 = *(const v2f*)(xrow + 2 * half);          // K = c in [0,4)
        v2f A1 = *(const v2f*)(xrow + 4 + 2 * half);      // K = c in [4,8)

        // B operand (4x16 f32): lane = N = d, VGPR pair = K={2*half, 2*half+1}
        const float* wrow = Wa + ((size_t)ig * DIM + col) * IN_DIM + 2 * half;
        v2f B0 = *(const v2f*)(wrow);
        v2f B1 = *(const v2f*)(wrow + 4);
        if (ib + t + IBLK < SUB)   // stream W ahead (global_prefetch_b8)
          __builtin_prefetch(Wa + ((size_t)(ig + IBLK) * DIM + col) * IN_DIM, 0, 1);

        v8f acc = {0.f, 0.f, 0.f, 0.f, 0.f, 0.f, 0.f, 0.f};
        acc = __builtin_amdgcn_wmma_f32_16x16x4_f32(false, A0, false, B0, (short)0, acc, false, false);
        acc = __builtin_amdgcn_wmma_f32_16x16x4_f32(false, A1, false, B1, (short)0, acc, false, false);
        u[t] = acc;

        // logits[b,a] = sum_d u * Pvec : DPP rotate-reduce over the 16-lane row
        float lj[8];
        #pragma unroll
        for (int j = 0; j < 8; ++j) lj[j] = rowsum16(u[t][j] * pv[j]);
        if (col == 0) {      // lanes 0 and 16 publish rows b=j+8*half
          #pragma unroll
          for (int j = 0; j < 8; ++j) lg[p][t][a][j + 8 * half] = lj[j];
        }
      }
      __syncthreads();

      // ---- Phase B: softmax over capsules + weighted accumulation
      #pragma unroll
      for (int t = 0; t < IBLK; ++t) {
        // softmax over capsules a for row b = col (replicated in both halves)
        float rd[16];
        #pragma unroll
        for (int k = 0; k < 16; ++k) rd[k] = lg[p][t][k][col];
        float mx = rd[0];
        #pragma unroll
        for (int k = 1; k < 16; ++k) mx = fmaxf(mx, rd[k]);
        float den = 0.f, va = 0.f;
        #pragma unroll
        for (int k = 0; k < 16; ++k) {
          float e = __expf(rd[k] - mx);
          den += e;
          va = (k == a) ? e : va;     // a is wave-uniform
        }
        float cc = va / den;          // c[b=col, a]

        // s[b,a,d] += c[b,a] * u[b,a,i,d] : broadcast c to each accumulator row
        #pragma unroll
        for (int j = 0; j < 8; ++j) {
          float cb = __shfl(cc, j + 8 * half, 32);
          sacc[j] = fmaf(cb, u[t][j], sacc[j]);
        }
      }
      p ^= 1;
    }
  }

  // write per-chunk partial s (deterministic, no atomics)
  #pragma unroll
  for (int j = 0; j < 8; ++j) {
    int bg = bt * 16 + j + 8 * half;
    partial[(((size_t)ci * B_TOT + bg) * CAPS + a) * DIM + col] = sacc[j];
  }
}

// Reduce partials, squash, accumulate routing state Pvec += in_caps * v,
// and on the final iteration emit v to d_out.
__global__ void __launch_bounds__(256) caps_squash(
    const float* __restrict__ partial, float* __restrict__ pvec,
    float* __restrict__ out, int NI, int final_iter)
{
  const int b = blockIdx.x;
  const int t = threadIdx.x;
  const int a = t >> 4;
  const int d = t & 15;
  float s = 0.f;
  for (int ci = 0; ci < NI; ++ci)
    s += partial[(((size_t)ci * B_TOT + b) * CAPS + a) * DIM + d];
  float ss = rowsum16(s * s);                 // sum over d (16-lane row)
  float coef = (ss / (1.f + ss)) * rsqrtf(ss + EPS);
  float v = s * coef;
  int idx = (b * CAPS + a) * DIM + d;
  pvec[idx] += (float)IN_CAPS * v;
  if (final_iter) out[idx] = v;
}

extern "C" void kernel_launch(void* const* d_in, const int* in_sizes, int n_in,
                              void* d_out, int out_size, void* d_ws, size_t ws_size,
                              hipStream_t stream) {
  const float* x = (const float*)d_in[0];   // [256, 4096, 8] f32
  const float* W = (const float*)d_in[1];   // [16, 4096, 16, 8] f32
  float* out     = (float*)d_out;           // [256, 16, 16] f32

  float* pvec    = (float*)d_ws;                        // [256,16,16]
  float* partial = pvec + (size_t)B_TOT * CAPS * DIM;   // [NI,256,16,16]
  const size_t base = (size_t)B_TOT * CAPS * DIM * sizeof(float);

  // Pick largest chunk count that fits the workspace (deterministic given ws_size).
  int NI = 1;
  const int cands[6] = {32, 16, 8, 4, 2, 1};
  for (int k = 0; k < 6; ++k) {
    size_t need = base + (size_t)cands[k] * B_TOT * CAPS * DIM * sizeof(float);
    if (need <= ws_size) { NI = cands[k]; break; }
  }

  init_pvec<<<(B_TOT * CAPS * DIM + 255) / 256, 256, 0, stream>>>(pvec);
  for (int t = 0; t < ROUTING_ITERS; ++t) {
    dim3 grid(NI, B_TOT / 16);
    caps_route_accum<<<grid, 512, 0, stream>>>(x, W, pvec, partial, NI);
    caps_squash<<<B_TOT, 256, 0, stream>>>(partial, pvec, out, NI,
                                           (t == ROUTING_ITERS - 1) ? 1 : 0);
  }
}